// DeepSeekV2Attention_1563368096321
// MI455X (gfx1250) — compile-verified
//
#include <hip/hip_runtime.h>
#include <cstddef>
#include <cstdint>

// ---------------------------------------------------------------------------
// DeepSeek-V2 MLA attention, fp32, CDNA5 (gfx1250) wave32 + V_WMMA_F32_16X16X4_F32
// ---------------------------------------------------------------------------

typedef __attribute__((ext_vector_type(2))) float v2f;
typedef __attribute__((ext_vector_type(4))) float v4f;
typedef __attribute__((ext_vector_type(8))) float v8f;

#define B_DIM  2
#define S_DIM  2048
#define HID    2048
#define H_DIM  16
#define D_NOPE 128
#define ATT_SCALE 0.08838834764831845f  // 128^-0.5

__device__ __forceinline__ v8f wmma_f32(v2f a, v2f b, v8f c) {
  // D = A(16x4 f32) * B(4x16 f32) + C(16x16 f32)
  return __builtin_amdgcn_wmma_f32_16x16x4_f32(false, a, false, b, (short)0, c,
                                               false, false);
}

// ---------------------------------------------------------------------------
// GEMM: C[M,N] = A[M,K] @ W[N,K]^T   (K == 2048 here, multiples of 16)
// MODE 0: plain row-major store to out0[M,N]
// MODE 1: q_nope: weight row remapped (skip rope rows), store (B,H,S,128)
// MODE 2: kv split: n<128 -> out0 (k), n>=128 -> out1 (v), each [M,128]
// Workgroup: 256 threads / 8 waves, 128x128 C tile (~32 FLOP/byte from cache),
// K-tiles of 16 staged in LDS; next K-tile prefetched while computing.
// ---------------------------------------------------------------------------
template <int MODE>
__global__ __launch_bounds__(256) void gemm_wt(const float* __restrict__ A,
                                               const float* __restrict__ W,
                                               float* __restrict__ out0,
                                               float* __restrict__ out1,
                                               int M, int N, int K) {
  __shared__ float As[128][20];  // pad 16->20: conflict-free frag reads, 16B rows
  __shared__ float Bs[128][20];

  const int tid  = threadIdx.x;
  const int wave = tid >> 5;
  const int lane = tid & 31;
  const int l    = lane & 15;
  const int hs   = lane >> 4;  // half-wave select (K offset +2 for lanes 16..31)
  const int m0   = blockIdx.y * 128;
  const int n0   = blockIdx.x * 128;

  v8f acc[8] = {};

  for (int kt = 0; kt < K; kt += 16) {
    __syncthreads();
    // Cooperative tile load: 128x16 floats each for A and B, 2 float4 per thread
#pragma unroll
    for (int i = 0; i < 2; ++i) {
      int ff  = tid * 2 + i;      // float4 index 0..511
      int row = ff >> 2;
      int c   = (ff & 3) * 4;
      const float* ap = A + (size_t)(m0 + row) * K + kt + c;
      v4f av = *(const v4f*)ap;
      *(v4f*)&As[row][c] = av;
      int wn = n0 + row;
      int wr = (MODE == 1) ? ((wn >> 7) * 192 + (wn & 127)) : wn;  // skip rope rows
      const float* bp = W + (size_t)wr * K + kt + c;
      v4f bv = *(const v4f*)bp;
      *(v4f*)&Bs[row][c] = bv;
      if (kt + 16 < K) {  // prefetch next K-tile (global_prefetch_b8)
        __builtin_prefetch(ap + 16, 0, 0);
        __builtin_prefetch(bp + 16, 0, 0);
      }
    }
    __syncthreads();
#pragma unroll
    for (int k0 = 0; k0 < 16; k0 += 4) {
      const int ak = k0 + 2 * hs;
      v2f a = *(const v2f*)&As[wave * 16 + l][ak];
#pragma unroll
      for (int nt = 0; nt < 8; ++nt) {
        v2f b = *(const v2f*)&Bs[nt * 16 + l][ak];
        acc[nt] = wmma_f32(a, b, acc[nt]);
      }
    }
  }

  // Epilogue. C layout: vgpr g, lanes0-15 -> M=g, lanes16-31 -> M=g+8; N=lane%16
#pragma unroll
  for (int nt = 0; nt < 8; ++nt) {
#pragma unroll
    for (int g = 0; g < 8; ++g) {
      int gm  = m0 + wave * 16 + hs * 8 + g;
      int gn  = n0 + nt * 16 + l;
      float v = acc[nt][g];
      if (MODE == 0) {
        out0[(size_t)gm * N + gn] = v;
      } else if (MODE == 1) {
        int h = gn >> 7, d = gn & 127;
        int b = gm >> 11, s = gm & 2047;  // M = B*S, S=2048
        out0[(((size_t)b * H_DIM + h) * S_DIM + s) * D_NOPE + d] = v;
      } else {
        if (gn < 128) out0[(size_t)gm * D_NOPE + gn] = v;
        else          out1[(size_t)gm * D_NOPE + (gn - 128)] = v;
      }
    }
  }
}

// ---------------------------------------------------------------------------
// Flash attention (causal, shared single-head K/V).
// Grid: (S/64, H, B). Block: 128 threads / 4 waves; wave w owns query rows
// [m0+16w, m0+16w+16). Q frags live in registers; 16-key K/V blocks staged
// in LDS; online softmax state per row; P restaged via wave-private LDS to
// convert WMMA C-layout -> A-layout for the P@V matmul.
// ---------------------------------------------------------------------------
__global__ __launch_bounds__(128) void mla_attn(const float* __restrict__ Q,   // (B,H,S,128)
                                                const float* __restrict__ Kc,  // (B,S,128)
                                                const float* __restrict__ Vc,  // (B,S,128)
                                                float* __restrict__ O) {       // (B,S,H*128)
  __shared__ float Kl[16][132];     // pad 128->132 (16B rows, spread banks)
  __shared__ float Vl[16][132];
  __shared__ float Pl[4][16][18];   // per-wave P tile, pad 16->18 (8B-aligned v2f)

  const int tid  = threadIdx.x;
  const int wave = tid >> 5;
  const int lane = tid & 31;
  const int l    = lane & 15;
  const int hs   = lane >> 4;
  const int m0   = blockIdx.x * 64;
  const int h    = blockIdx.y;
  const int b    = blockIdx.z;
  // mw is wave-uniform; pin it to an SGPR so causal guards become s_cbranch.
  const int mw   = __builtin_amdgcn_readfirstlane(m0 + wave * 16);

  const float* q  = Q + ((size_t)b * H_DIM + h) * S_DIM * D_NOPE;
  const float* kp = Kc + (size_t)b * S_DIM * D_NOPE;
  const float* vp = Vc + (size_t)b * S_DIM * D_NOPE;

  // Preload Q A-fragments (scaled by 1/sqrt(d)): 32 k-steps of 4
  v2f qf[32];
  {
    const float* qr = q + (size_t)(mw + l) * D_NOPE + 2 * hs;
#pragma unroll
    for (int kk = 0; kk < 32; ++kk) {
      v2f t = *(const v2f*)(qr + kk * 4);
      qf[kk].x = t.x * ATT_SCALE;
      qf[kk].y = t.y * ATT_SCALE;
    }
  }

  v8f oacc[8] = {};
  float mstate[8], lstate[8];
#pragma unroll
  for (int g = 0; g < 8; ++g) { mstate[g] = -__builtin_inff(); lstate[g] = 0.f; }

  const int kend = m0 + 64;  // keys needed by this workgroup (causal)
  for (int kb = 0; kb < kend; kb += 16) {
    __syncthreads();
    // Cooperative K/V block load: 16x128 floats each -> 4 float4 per thread each
#pragma unroll
    for (int i = 0; i < 4; ++i) {
      int ff  = tid * 4 + i;  // 0..511
      int row = ff >> 5;
      int c   = (ff & 31) * 4;
      *(v4f*)&Kl[row][c] = *(const v4f*)(kp + (size_t)(kb + row) * D_NOPE + c);
      *(v4f*)&Vl[row][c] = *(const v4f*)(vp + (size_t)(kb + row) * D_NOPE + c);
    }
    __syncthreads();

    const bool active = (kb <= mw + 15);  // scalar (mw in SGPR): s_cbranch, EXEC untouched
    if (active) {
      // S = Q @ K^T (16x16), 32 WMMA steps over d=128
      v8f s = {};
#pragma unroll
      for (int kk = 0; kk < 32; ++kk) {
        v2f kb2 = *(const v2f*)&Kl[l][kk * 4 + 2 * hs];  // B[k][n]=K[kb+n][k]
        s = wmma_f32(qf[kk], kb2, s);
      }
      // Causal mask + online softmax (row-wise over 16-lane halves)
#pragma unroll
      for (int g = 0; g < 8; ++g) {
        float sv = s[g];
        int qi = mw + hs * 8 + g;
        int kj = kb + l;
        if (kj > qi) sv = -__builtin_inff();
        float rmax = sv;
#pragma unroll
        for (int off = 1; off < 16; off <<= 1)
          rmax = fmaxf(rmax, __shfl_xor(rmax, off, 32));
        float mnew  = fmaxf(mstate[g], rmax);
        float alpha = (mnew == -__builtin_inff()) ? 1.f : __expf(mstate[g] - mnew);
        float p     = (mnew == -__builtin_inff()) ? 0.f : __expf(sv - mnew);
        float rsum = p;
#pragma unroll
        for (int off = 1; off < 16; off <<= 1)
          rsum += __shfl_xor(rsum, off, 32);
        lstate[g] = lstate[g] * alpha + rsum;
        mstate[g] = mnew;
#pragma unroll
        for (int nt = 0; nt < 8; ++nt) oacc[nt][g] *= alpha;
        Pl[wave][hs * 8 + g][l] = p;  // C-layout element (m, n)
      }
    }
    __syncthreads();  // publish P (layout conversion through LDS)
    if (active) {
      // O += P(16x16) @ V(16x128)
#pragma unroll
      for (int k0 = 0; k0 < 16; k0 += 4) {
        v2f pa = *(const v2f*)&Pl[wave][l][k0 + 2 * hs];  // A-layout read
#pragma unroll
        for (int nt = 0; nt < 8; ++nt) {
          v2f vb;
          vb.x = Vl[k0 + 2 * hs][nt * 16 + l];      // B[k][n]=V[kb+k][n]
          vb.y = Vl[k0 + 2 * hs + 1][nt * 16 + l];
          oacc[nt] = wmma_f32(pa, vb, oacc[nt]);
        }
      }
    }
  }

  // Epilogue: O * (1/l)  ->  attn_out (B,S,H*128)
#pragma unroll
  for (int g = 0; g < 8; ++g) {
    float inv = 1.0f / lstate[g];
    int gm = mw + hs * 8 + g;
#pragma unroll
    for (int nt = 0; nt < 8; ++nt) {
      int gn = h * D_NOPE + nt * 16 + l;
      O[((size_t)b * S_DIM + gm) * HID + gn] = oacc[nt][g] * inv;
    }
  }
}

// ---------------------------------------------------------------------------
extern "C" void kernel_launch(void* const* d_in, const int* in_sizes, int n_in,
                              void* d_out, int out_size, void* d_ws, size_t ws_size,
                              hipStream_t stream) {
  const float* hidden = (const float*)d_in[0];  // (B,S,HID)
  const float* Wq     = (const float*)d_in[1];  // (H*(128+64), HID)
  const float* Wkv    = (const float*)d_in[2];  // (256, HID)
  const float* Wo     = (const float*)d_in[3];  // (HID, H*128)
  float* out = (float*)d_out;                   // (B,S,HID)

  // Workspace layout (floats): q 8M | k 0.5M | v 0.5M | attn_out 8M  (~68 MB)
  float* qws = (float*)d_ws;
  float* kws = qws + (size_t)B_DIM * H_DIM * S_DIM * D_NOPE;
  float* vws = kws + (size_t)B_DIM * S_DIM * D_NOPE;
  float* aws = vws + (size_t)B_DIM * S_DIM * D_NOPE;

  const int M = B_DIM * S_DIM;  // 4096

  // 1) q_nope = hidden @ Wq_nope^T  -> (B,H,S,128)
  gemm_wt<1><<<dim3((H_DIM * D_NOPE) / 128, M / 128), 256, 0, stream>>>(
      hidden, Wq, qws, nullptr, M, H_DIM * D_NOPE, HID);
  // 2) kv = hidden @ Wkv^T[:256] -> k (B,S,128), v (B,S,128)
  gemm_wt<2><<<dim3(256 / 128, M / 128), 256, 0, stream>>>(
      hidden, Wkv, kws, vws, M, 256, HID);
  // 3) causal flash attention -> (B,S,H*128)
  mla_attn<<<dim3(S_DIM / 64, H_DIM, B_DIM), 128, 0, stream>>>(qws, kws, vws, aws);
  // 4) out = attn_out @ Wo^T
  gemm_wt<0><<<dim3(HID / 128, M / 128), 256, 0, stream>>>(
      aws, Wo, out, nullptr, M, HID, HID);
}